// FlashChatGLM3Attention_65541200937057
// MI455X (gfx1250) — compile-verified
//
#include <hip/hip_runtime.h>
#include <math.h>

typedef __attribute__((ext_vector_type(2))) float v2f;
typedef __attribute__((ext_vector_type(4))) float v4f;
typedef __attribute__((ext_vector_type(8))) float v8f;
typedef __attribute__((ext_vector_type(4))) unsigned int v4u;
typedef __attribute__((ext_vector_type(4))) int v4i;
typedef __attribute__((ext_vector_type(8))) int v8i;

#define HIDDEN   4096
#define NQH      32
#define NKVH     2
#define HD       128
#define PRE_B    2
#define PRE_L    1024
#define PRE_TOT  2048
#define DEC_B    16
#define PAST     4096
#define TOT      2064
#define QKV_OUT  4608
#define QDIM     4096
#define VOFF     4352          /* 4096 q + 256 k */
#define ATT_SCALE 0.08838834764831845f  /* 1/sqrt(128) */
#define LOG1E4_64 0.14391156724f        /* ln(10000)/64 */

#if defined(__has_builtin)
#if __has_builtin(__builtin_amdgcn_tensor_load_to_lds) && \
    __has_builtin(__builtin_amdgcn_s_wait_tensorcnt)
#define USE_TDM 1
#endif
#endif
#ifndef USE_TDM
#define USE_TDM 0
#endif

__device__ __forceinline__ v8f wmma_f32(v2f a, v2f b, v8f c) {
    // V_WMMA_F32_16X16X4_F32 : D = A(16x4) * B(4x16) + C(16x16), full f32
    return __builtin_amdgcn_wmma_f32_16x16x4_f32(
        false, a, false, b, (short)0, c, false, false);
}

#if USE_TDM
// TDM 2-D tile load: `rows` rows of 128 f32, global row stride = QKV_OUT f32,
// LDS row stride = 132 f32 (4-dword pad per 128-dword row -> bank stagger).
// D# per CDNA5 ISA ch.8: group0 {count=1, lds_addr, global_addr, type=2},
// group1 {data_size=4B, pad_en, pad_interval=128dw, pad_amount=4dw,
//         tensor_dim0=128, tensor_dim1=1M, tile=128x rows, dim0_stride=4608}.
__device__ __forceinline__ void tdm_load_v_tile(const float* gsrc,
                                                unsigned lds_off, int rows)
{
    unsigned long long ga = (unsigned long long)(__SIZE_TYPE__)gsrc;
    v4u g0;
    g0.x = 1u;                                          // count=1, load, user
    g0.y = lds_off;                                     // LDS byte address
    g0.z = (unsigned)(ga & 0xFFFFFFFFu);                // global_addr[31:0]
    g0.w = (unsigned)((ga >> 32) & 0x01FFFFFFu) | (2u << 30); // [56:32]|type=2

    const unsigned td0 = 128u;          // tensor dim0 (f32 units)
    const unsigned td1 = 1u << 20;      // tensor dim1 (large, OOB never hit)
    v8i g1;
    g1[0] = (int)((2u << 16)            // data_size = 4 bytes
                | (1u << 20)            // pad_enable
                | (6u << 22)            // pad_interval: 128 dwords
                | (3u << 25));          // pad_amount:   4 dwords
    g1[1] = (int)((td0 & 0xFFFFu) << 16);               // tensor_dim0[15:0]
    g1[2] = (int)(((td0 >> 16) & 0xFFFFu) | ((td1 & 0xFFFFu) << 16));
    g1[3] = (int)(((td1 >> 16) & 0xFFFFu) | (128u << 16)); // tile_dim0 = 128
    g1[4] = (int)((unsigned)rows);                      // tile_dim1, tile_dim2=0
    g1[5] = (int)(unsigned)QKV_OUT;                     // dim0_stride[31:0]
    g1[6] = 0;
    g1[7] = 0;
    v4i gz4 = {0, 0, 0, 0};
    v8i gz8 = {0, 0, 0, 0, 0, 0, 0, 0};
    __builtin_amdgcn_tensor_load_to_lds(g0, g1, gz4, gz4, gz8, 0);
}
#endif

// 16-wide K-chunk of register fragments (1 A frag + 4 B frags per k-step of 4)
struct KChunk {
    v2f a[4];
    v2f b[4][4];
};

// ---------------------------------------------------------------------------
// C[M,N] = A[M,K] @ W[N,K]^T (+ bias). NT GEMM, K-contiguous both sides.
// 256 thr = 8 waves (2 M x 4 N). Tile 32(M) x 256(N); wave tile 16x64.
// K stepped by 32 with two *named* fragment sets (no dynamic buffer index ->
// no movrel / scratch spills): compute(f0) overlaps load(f1) and vice versa.
// ---------------------------------------------------------------------------
__global__ __launch_bounds__(256)
void gemm_nt_wmma_kernel(const float* __restrict__ A,
                         const float* __restrict__ W,
                         const float* __restrict__ bias,
                         float* __restrict__ C,
                         int M, int N, int K)
{
    const int tid    = threadIdx.x;
    const int wave   = tid >> 5;
    const int lane   = tid & 31;
    const int laneLo = lane & 15;
    const int laneHi = lane >> 4;
    const int wM     = wave & 1;
    const int wN     = wave >> 1;
    const int blockM = blockIdx.x * 32;
    const int blockN = blockIdx.y * 256;

    int mrow  = blockM + wM * 16 + laneLo;
    int mload = (mrow < M) ? mrow : (M - 1);   // clamp -> EXEC stays uniform
    const float* __restrict__ arow = A + (size_t)mload * K;

    const float* wrow[4];
    int ncol[4];
#pragma unroll
    for (int nt = 0; nt < 4; ++nt) {
        ncol[nt] = blockN + wN * 64 + nt * 16 + laneLo;
        wrow[nt] = W + (size_t)ncol[nt] * K;
    }

    v8f acc[4] = {};

    auto load_chunk = [&](int k, KChunk& f) {
#pragma unroll
        for (int kk = 0; kk < 4; ++kk) {
            const int kb = k + kk * 4 + 2 * laneHi;
            f.a[kk] = *(const v2f*)(arow + kb);
#pragma unroll
            for (int nt = 0; nt < 4; ++nt)
                f.b[kk][nt] = *(const v2f*)(wrow[nt] + kb);
        }
    };
    auto compute_chunk = [&](const KChunk& f) {
#pragma unroll
        for (int kk = 0; kk < 4; ++kk)
#pragma unroll
            for (int nt = 0; nt < 4; ++nt)
                acc[nt] = wmma_f32(f.a[kk], f.b[kk][nt], acc[nt]);
    };

    KChunk f0, f1;
    load_chunk(0, f0);
    for (int k = 0; k < K; k += 32) {          // K % 32 == 0 for all our GEMMs
        load_chunk(k + 16, f1);
        __builtin_prefetch((const void*)(wrow[0] + k + 544), 0, 3);
        compute_chunk(f0);
        if (k + 32 < K) load_chunk(k + 32, f0);
        compute_chunk(f1);
    }

    // D layout: VGPR r -> row (r + 8*laneHi), col = laneLo
#pragma unroll
    for (int nt = 0; nt < 4; ++nt) {
        const int col = ncol[nt];
        const float bv = bias ? bias[col] : 0.0f;
#pragma unroll
        for (int r = 0; r < 8; ++r) {
            const int row = blockM + wM * 16 + r + 8 * laneHi;
            if (row < M) C[(size_t)row * N + col] = acc[nt][r] + bv;
        }
    }
}

// ---------------------------------------------------------------------------
// RoPE in-place. grid = (token, head 0..33), 64 threads (one freq pair each).
// ---------------------------------------------------------------------------
__global__ __launch_bounds__(64)
void rope_kernel(float* __restrict__ qkv)
{
    const int t  = blockIdx.x;
    const int hh = blockIdx.y;
    const int i  = threadIdx.x;
    const int pos = (t < PRE_TOT) ? (t & (PRE_L - 1)) : PAST;

    const float inv_freq = expf(-(float)i * LOG1E4_64);
    float s, c;
    sincosf((float)pos * inv_freq, &s, &c);

    float* base;
    if (hh < NQH) base = qkv + (size_t)t * QKV_OUT + hh * HD;
    else          base = qkv + (size_t)t * QKV_OUT + QDIM + (hh - NQH) * HD;

    const float x1 = base[i];
    const float x2 = base[i + 64];
    base[i]      = x1 * c - x2 * s;
    base[i + 64] = x2 * c + x1 * s;
}

// ---------------------------------------------------------------------------
// Prefill attention: WG per (16-query block, head, batch). 8 waves.
// QK^T on WMMA -> 16x1024 score tile in LDS -> softmax -> PV on WMMA with
// V tiles staged to LDS by the Tensor Data Mover (64 rows per chunk).
// ---------------------------------------------------------------------------
#define SSTR 1032   /* score row stride: 1032 % 64 = 8 -> bank stagger */
#define VSTR 132    /* V tile row stride: 128 data + 4 pad dwords (TDM pad) */

__global__ __launch_bounds__(256)
void prefill_attn_kernel(const float* __restrict__ qkv,
                         float* __restrict__ attn)
{
    __shared__ __align__(16) float sS[16 * SSTR];   // 66 KB scores/probs
    __shared__ __align__(16) float sV[64 * VSTR];   // 33 KB V tile
    __shared__ float sRed[16 * 16];
    __shared__ float sSum[16];

    const int rb  = blockIdx.x;
    const int h   = blockIdx.y;
    const int b   = blockIdx.z;
    const int kvh = h >> 4;                 // GQA 16:1

    const int tid    = threadIdx.x;
    const int wave   = tid >> 5;
    const int lane   = tid & 31;
    const int laneLo = lane & 15;
    const int laneHi = lane >> 4;

    const int q0 = rb * 16;
    const size_t tokbase = (size_t)b * PRE_L;

    // ---- preload Q A-fragments (reused by all 8 key tiles of this wave)
    const float* __restrict__ qrow =
        qkv + (tokbase + q0 + laneLo) * (size_t)QKV_OUT + (size_t)h * HD;
    v2f afrag[32];
#pragma unroll
    for (int d4 = 0; d4 < 32; ++d4)
        afrag[d4] = *(const v2f*)(qrow + d4 * 4 + 2 * laneHi);

    // ---- S = Q K^T * scale, causal-masked. wave owns cols [w*128, w*128+128)
    for (int nt = 0; nt < 8; ++nt) {
        const int col = wave * 128 + nt * 16 + laneLo;
        const float* __restrict__ krow =
            qkv + (tokbase + col) * (size_t)QKV_OUT + QDIM + (size_t)kvh * HD;
        v2f bfrag[32];
#pragma unroll
        for (int d4 = 0; d4 < 32; ++d4)          // load burst (clustered)
            bfrag[d4] = *(const v2f*)(krow + d4 * 4 + 2 * laneHi);
        v8f acc = {};
#pragma unroll
        for (int d4 = 0; d4 < 32; ++d4)          // WMMA burst
            acc = wmma_f32(afrag[d4], bfrag[d4], acc);
#pragma unroll
        for (int r = 0; r < 8; ++r) {
            const int row  = r + 8 * laneHi;
            float v = acc[r] * ATT_SCALE;
            if (col > q0 + row) v = -1e30f;
            sS[row * SSTR + col] = v;
        }
    }
    __syncthreads();

    // ---- softmax: row = tid&15, 16 slots cooperate per row
    {
        const int row = tid & 15, slot = tid >> 4;
        float m = -1e30f;
        for (int c2 = slot; c2 < PRE_L; c2 += 16)
            m = fmaxf(m, sS[row * SSTR + c2]);
        sRed[row * 16 + slot] = m;
        __syncthreads();
        float rm = -1e30f;
#pragma unroll
        for (int s2 = 0; s2 < 16; ++s2) rm = fmaxf(rm, sRed[row * 16 + s2]);

        float psum = 0.f;
        for (int c2 = slot; c2 < PRE_L; c2 += 16) {
            const float e = __expf(sS[row * SSTR + c2] - rm);
            sS[row * SSTR + c2] = e;
            psum += e;
        }
        __syncthreads();
        sRed[row * 16 + slot] = psum;
        __syncthreads();
        if (slot == 0) {
            float tot = 0.f;
#pragma unroll
            for (int s2 = 0; s2 < 16; ++s2) tot += sRed[row * 16 + s2];
            sSum[row] = tot;
        }
        __syncthreads();
    }

    // ---- O = P V.  V staged in 64-row LDS tiles (TDM), wave owns 16 dims.
    const int dcol = wave * 16 + laneLo;
    const float* __restrict__ vbase =
        qkv + tokbase * QKV_OUT + VOFF + (size_t)kvh * HD;
#if USE_TDM
    const unsigned sv_off = (unsigned)((__SIZE_TYPE__)(void*)sV & 0xFFFFFFFFu);
#endif
    v8f acc = {};
    for (int chunk = 0; chunk < PRE_L; chunk += 64) {
#if USE_TDM
        if (wave == 0) {
            tdm_load_v_tile(vbase + (size_t)chunk * QKV_OUT, sv_off, 64);
            __builtin_amdgcn_s_wait_tensorcnt(0);
        }
#else
        for (int i = tid; i < (64 * 128) / 4; i += 256) {
            const int idx = i * 4;
            const int r = idx >> 7, c = idx & 127;
            *(v4f*)&sV[r * VSTR + c] =
                *(const v4f*)(vbase + (size_t)(chunk + r) * QKV_OUT + c);
        }
#endif
        __syncthreads();
#pragma unroll
        for (int k = 0; k < 64; k += 4) {
            const int kb = k + 2 * laneHi;
            v2f a = *(const v2f*)&sS[laneLo * SSTR + chunk + kb];
            v2f bf;
            bf.x = sV[kb * VSTR + dcol];
            bf.y = sV[(kb + 1) * VSTR + dcol];
            acc = wmma_f32(a, bf, acc);
        }
        __syncthreads();
    }
#pragma unroll
    for (int r = 0; r < 8; ++r) {
        const int row   = r + 8 * laneHi;
        const int token = (int)tokbase + q0 + row;
        attn[(size_t)token * HIDDEN + (size_t)h * HD + dcol] = acc[r] / sSum[row];
    }
}

// ---------------------------------------------------------------------------
// Decode attention: 1 WG per (head, batch), 128 threads. On-the-fly RoPE of
// cached K (freq table in LDS); new-token K/Q already roped in qkv buffer.
// ---------------------------------------------------------------------------
__global__ __launch_bounds__(128)
void decode_attn_kernel(const float* __restrict__ qkv,
                        const float* __restrict__ cacheK,
                        const float* __restrict__ cacheV,
                        float* __restrict__ attn)
{
    __shared__ float sQ[HD];
    __shared__ float sInv[64];
    __shared__ float sScore[PAST + 1];
    __shared__ float sRed[128];

    const int h   = blockIdx.x;
    const int b   = blockIdx.y;
    const int kvh = h >> 4;
    const int tid = threadIdx.x;
    const int dect = PRE_TOT + b;

    sQ[tid] = qkv[(size_t)dect * QKV_OUT + (size_t)h * HD + tid];
    if (tid < 64) sInv[tid] = expf(-(float)tid * LOG1E4_64);
    __syncthreads();

    for (int p = tid; p <= PAST; p += 128) {
        float dot = 0.f;
        if (p == PAST) {
            const float* kr = qkv + (size_t)dect * QKV_OUT + QDIM + (size_t)kvh * HD;
            for (int d = 0; d < HD; ++d) dot += sQ[d] * kr[d];
        } else {
            const float* kc = cacheK + (((size_t)b * PAST + p) * NKVH + kvh) * HD;
            for (int i = 0; i < 64; ++i) {
                float s, c;
                sincosf((float)p * sInv[i], &s, &c);
                const float x1 = kc[i], x2 = kc[i + 64];
                dot += sQ[i] * (x1 * c - x2 * s) + sQ[i + 64] * (x2 * c + x1 * s);
            }
        }
        sScore[p] = dot * ATT_SCALE;
    }
    __syncthreads();

    float m = -1e30f;
    for (int p = tid; p <= PAST; p += 128) m = fmaxf(m, sScore[p]);
    sRed[tid] = m; __syncthreads();
    for (int off = 64; off > 0; off >>= 1) {
        if (tid < off) sRed[tid] = fmaxf(sRed[tid], sRed[tid + off]);
        __syncthreads();
    }
    const float rm = sRed[0];
    __syncthreads();

    float psum = 0.f;
    for (int p = tid; p <= PAST; p += 128) {
        const float e = __expf(sScore[p] - rm);
        sScore[p] = e;
        psum += e;
    }
    sRed[tid] = psum; __syncthreads();
    for (int off = 64; off > 0; off >>= 1) {
        if (tid < off) sRed[tid] += sRed[tid + off];
        __syncthreads();
    }
    const float tot = sRed[0];
    __syncthreads();

    float acc = 0.f;
    const float* __restrict__ vcol =
        cacheV + ((size_t)b * PAST * NKVH + kvh) * HD + tid;
    for (int p = 0; p < PAST; ++p)
        acc += sScore[p] * vcol[(size_t)p * (NKVH * HD)];
    acc += sScore[PAST] * qkv[(size_t)dect * QKV_OUT + VOFF + (size_t)kvh * HD + tid];

    attn[(size_t)dect * HIDDEN + (size_t)h * HD + tid] = acc / tot;
}

// ---------------------------------------------------------------------------
extern "C" void kernel_launch(void* const* d_in, const int* in_sizes, int n_in,
                              void* d_out, int out_size, void* d_ws, size_t ws_size,
                              hipStream_t stream)
{
    const float* hidden = (const float*)d_in[0];
    const float* Wqkv   = (const float*)d_in[1];
    const float* bqkv   = (const float*)d_in[2];
    const float* Wo     = (const float*)d_in[3];
    const float* cacheK = (const float*)d_in[4];
    const float* cacheV = (const float*)d_in[5];
    float* out  = (float*)d_out;

    float* qkv  = (float*)d_ws;                      // [2064, 4608]
    float* attn = qkv + (size_t)TOT * QKV_OUT;       // [2064, 4096]

    dim3 g1((TOT + 31) / 32, QKV_OUT / 256);
    gemm_nt_wmma_kernel<<<g1, 256, 0, stream>>>(hidden, Wqkv, bqkv, qkv,
                                                TOT, QKV_OUT, HIDDEN);

    dim3 g2(TOT, NQH + NKVH);
    rope_kernel<<<g2, 64, 0, stream>>>(qkv);

    dim3 g3(PRE_L / 16, NQH, PRE_B);
    prefill_attn_kernel<<<g3, 256, 0, stream>>>(qkv, attn);

    dim3 g4(NQH, DEC_B);
    decode_attn_kernel<<<g4, 128, 0, stream>>>(qkv, cacheK, cacheV, attn);

    dim3 g5((TOT + 31) / 32, HIDDEN / 256);
    gemm_nt_wmma_kernel<<<g5, 256, 0, stream>>>(attn, Wo, nullptr, out,
                                                TOT, HIDDEN, HIDDEN);
}